// LinearSelfAttention_64252710748630
// MI455X (gfx1250) — compile-verified
//
#include <hip/hip_runtime.h>
#include <hip/hip_bf16.h>

// LinearSelfAttention (MobileViT-style) on MI455X / gfx1250.
//
// Shapes: x (16,512,4,1024) f32; w_qkv (1025,512); b_qkv (1025); w_out (512,512); b_out (512).
// Token = (b,p,n), 65536 tokens; group g=(b,p), 64 groups of N=1024 tokens.
//
// Algebraic fusion: context_vector = W_k @ xbar + b_k where
//   xbar[g,:] = sum_n softmax(q)[g,n] * x[g,:,n]   (K GEMM eliminated, exact in real arith).
// Heavy path: out_t = W_out @ (relu(W_v @ x_t + b_v) * cv[g]) + b_out  -> fused WMMA kernel
// with 64-token tiles (256 KB LDS, CDNA5 WGP has 320 KB) and 4-way A-fragment register reuse.

#define C_DIM   512
#define N_DIM   1024
#define P_DIM   4
#define CPN     (C_DIM * P_DIM * N_DIM)   // per-b stride in x/out
#define PN      (P_DIM * N_DIM)           // per-c stride
#define TTOK    64                        // tokens per block in main kernel
#define NTILE   4                         // 16-token sub-tiles per block
#define PANEL   (C_DIM * 16)              // floats per [c][n16] LDS panel (8192)

typedef __attribute__((ext_vector_type(2))) float v2f;
typedef __attribute__((ext_vector_type(8))) float v8f;

// workspace layout (float offsets)
#define WS_Q      0          // 65536
#define WS_SCORE  65536      // 65536
#define WS_XBAR   131072     // 64*512
#define WS_CV     163840     // 64*512
#define WS_WVPK   196608     // 512*512 packed A-fragments for W_v
#define WS_WOPK   458752     // 512*512 packed A-fragments for W_out

// ---------------------------------------------------------------------------
// Pack W_v (rows 513..1024 of w_qkv) and W_out into WMMA A-fragment order:
// fragment element v (0..1) for (kstep, mtile, lane):
//   m = mtile*16 + lane%16,  k = kstep*4 + v + 2*(lane/16)
// stored at float offset 2*((kstep*32 + mtile)*32 + lane) + v
// so the main kernel's A load is one coalesced b64 per wave.
__global__ void pack_w_kernel(const float* __restrict__ w_qkv,
                              const float* __restrict__ w_out,
                              float* __restrict__ wvpk, float* __restrict__ wopk) {
    int idx   = blockIdx.x * blockDim.x + threadIdx.x;   // 0 .. 131071
    int lane  = idx & 31;
    int mtile = (idx >> 5) & 31;
    int kstep = idx >> 10;
    int m     = mtile * 16 + (lane & 15);
    int kb    = kstep * 4 + 2 * (lane >> 4);
    const float* wv = w_qkv + (1 + C_DIM) * C_DIM;       // value rows start at 513
    wvpk[2 * idx]     = wv[m * C_DIM + kb];
    wvpk[2 * idx + 1] = wv[m * C_DIM + kb + 1];
    wopk[2 * idx]     = w_out[m * C_DIM + kb];
    wopk[2 * idx + 1] = w_out[m * C_DIM + kb + 1];
}

// ---------------------------------------------------------------------------
// q[g,n] = w_qkv[0,:] . x[b,:,p,n] + b_qkv[0]   (one thread per token, coalesced over n)
__global__ void q_kernel(const float* __restrict__ x, const float* __restrict__ w_qkv,
                         const float* __restrict__ b_qkv, float* __restrict__ q) {
    int idx = blockIdx.x * blockDim.x + threadIdx.x;     // token id
    int g = idx >> 10, n = idx & 1023;
    int b = g >> 2,    p = g & 3;
    const float* xp = x + (size_t)b * CPN + p * N_DIM + n;
    float acc = b_qkv[0];
#pragma unroll 8
    for (int c = 0; c < C_DIM; ++c)
        acc = fmaf(w_qkv[c], xp[(size_t)c * PN], acc);
    q[idx] = acc;
}

// ---------------------------------------------------------------------------
// softmax over n (1024) per group; 64 blocks x 256 threads, 4 elems/thread
__global__ void softmax_kernel(const float* __restrict__ q, float* __restrict__ score) {
    __shared__ float red[256];
    int g = blockIdx.x, t = threadIdx.x;
    float v[4];
    float m = -__builtin_inff();
#pragma unroll
    for (int k = 0; k < 4; ++k) { v[k] = q[g * 1024 + t + k * 256]; m = fmaxf(m, v[k]); }
    red[t] = m; __syncthreads();
    for (int s = 128; s > 0; s >>= 1) { if (t < s) red[t] = fmaxf(red[t], red[t + s]); __syncthreads(); }
    m = red[0]; __syncthreads();
    float sum = 0.f;
#pragma unroll
    for (int k = 0; k < 4; ++k) { v[k] = expf(v[k] - m); sum += v[k]; }
    red[t] = sum; __syncthreads();
    for (int s = 128; s > 0; s >>= 1) { if (t < s) red[t] += red[t + s]; __syncthreads(); }
    float inv = 1.f / red[0];
#pragma unroll
    for (int k = 0; k < 4; ++k) score[g * 1024 + t + k * 256] = v[k] * inv;
}

// ---------------------------------------------------------------------------
// xbar[g,c] = sum_n score[g,n] * x[b,c,p,n]   (one wave32 per (g,c) row; coalesced)
__global__ void xbar_kernel(const float* __restrict__ x, const float* __restrict__ score,
                            float* __restrict__ xbar) {
    int wid  = (blockIdx.x * blockDim.x + threadIdx.x) >> 5;
    int lane = threadIdx.x & 31;
    int g = wid >> 9, c = wid & 511;
    int b = g >> 2,   p = g & 3;
    const float* xp = x + (size_t)b * CPN + (size_t)c * PN + p * N_DIM;
    const float* sp = score + g * 1024;
    float acc = 0.f;
    for (int n = lane; n < 1024; n += 32) acc = fmaf(xp[n], sp[n], acc);
    for (int off = 16; off; off >>= 1) acc += __shfl_down(acc, off, 32);
    if (lane == 0) xbar[g * 512 + c] = acc;
}

// ---------------------------------------------------------------------------
// cv[g,o] = w_qkv[1+o,:] . xbar[g,:] + b_qkv[1+o]   (one wave per output)
__global__ void cv_kernel(const float* __restrict__ w_qkv, const float* __restrict__ b_qkv,
                          const float* __restrict__ xbar, float* __restrict__ cv) {
    int wid  = (blockIdx.x * blockDim.x + threadIdx.x) >> 5;
    int lane = threadIdx.x & 31;
    int g = wid >> 9, o = wid & 511;
    const float* wk = w_qkv + (size_t)(1 + o) * C_DIM;
    const float* xb = xbar + g * 512;
    float acc = 0.f;
    for (int c = lane; c < C_DIM; c += 32) acc = fmaf(wk[c], xb[c], acc);
    for (int off = 16; off; off >>= 1) acc += __shfl_down(acc, off, 32);
    if (lane == 0) cv[g * 512 + o] = acc + b_qkv[1 + o];
}

// ---------------------------------------------------------------------------
// Fused main kernel: per block = one 64-token tile of one group (4 sub-tiles of 16).
//   stage 1: V = W_v @ xtile (WMMA f32 16x16x4, K=512) ; h = relu(V+b_v)*cv -> LDS
//   stage 2: out = W_out @ h + b_out                    (WMMA)
// Each wave owns 4 mtiles x 4 ntiles (128 accumulator VGPRs); one A-fragment b64 load
// feeds 4 WMMAs -> 4x less L2 weight traffic than a 16-token tile.
// LDS: two 128 KB tiles ([ntile][c][n16] panels, bank-conflict-free B reads) = 256 KB,
// exploiting CDNA5's 320 KB per-workgroup LDS.
// WMMA f32 16x16x4 layouts (wave32):
//   A 16x4: lane L: m=L%16; VGPR v: k = v + 2*(L/16)       (pre-packed in ws)
//   B 4x16: lane L: n=L%16; VGPR v: k = v + 2*(L/16)       (from LDS panels)
//   C/D:    lane L: n=L%16; VGPR v: m = v + 8*(L/16)
__global__ void __launch_bounds__(256) main_kernel(
    const float* __restrict__ x, const float* __restrict__ b_qkv,
    const float* __restrict__ b_out, const float* __restrict__ cv,
    const float* __restrict__ wvpk, const float* __restrict__ wopk,
    float* __restrict__ out) {
    extern __shared__ float smem[];
    float* xs = smem;                      // x tile, NTILE panels of [c][n16], 128 KB
    float* hs = smem + NTILE * PANEL;      // h tile, same layout, 128 KB

    int g  = blockIdx.x >> 4;              // 16 blocks per group (1024/64)
    int n0 = (blockIdx.x & 15) * TTOK;
    int b  = g >> 2, p = g & 3;
    const size_t xbase = (size_t)b * CPN + (size_t)p * N_DIM + n0;

    int t = threadIdx.x;
    // cooperative x-tile load: lanes sweep n (contiguous 128B per wave-load)
    for (int i = t; i < C_DIM * TTOK; i += 256) {
        int c = i >> 6, n = i & 63;
        xs[(n >> 4) * PANEL + c * 16 + (n & 15)] = x[xbase + (size_t)c * PN + n];
    }
    __syncthreads();

    int wave = t >> 5, lane = t & 31;
    int half = lane >> 4, nn = lane & 15;

    // ---- stage 1: V GEMM ----
    v8f acc[4][NTILE] = {};
    for (int ks = 0; ks < 128; ++ks) {
        int krow = (ks * 4 + 2 * half) * 16 + nn;
        v2f bb[NTILE];
#pragma unroll
        for (int nt = 0; nt < NTILE; ++nt) {
            bb[nt].x = xs[nt * PANEL + krow];
            bb[nt].y = xs[nt * PANEL + krow + 16];
        }
        const v2f* ap = (const v2f*)wvpk + ((size_t)ks * 32 + wave * 4) * 32 + lane;
#pragma unroll
        for (int mi = 0; mi < 4; ++mi) {
            v2f aa = ap[(size_t)mi * 32];
#pragma unroll
            for (int nt = 0; nt < NTILE; ++nt)
                acc[mi][nt] = __builtin_amdgcn_wmma_f32_16x16x4_f32(
                    false, aa, false, bb[nt], (short)0, acc[mi][nt], false, false);
        }
    }
    // epilogue 1: bias + relu + *cv -> LDS h panels
#pragma unroll
    for (int mi = 0; mi < 4; ++mi) {
        int m0 = (wave * 4 + mi) * 16;
#pragma unroll
        for (int v = 0; v < 8; ++v) {
            int m = m0 + v + 8 * half;
            float bias = b_qkv[513 + m];
            float scale = cv[g * 512 + m];
#pragma unroll
            for (int nt = 0; nt < NTILE; ++nt)
                hs[nt * PANEL + m * 16 + nn] = fmaxf(acc[mi][nt][v] + bias, 0.f) * scale;
        }
    }
    __syncthreads();

    // ---- stage 2: out GEMM ----
    v8f acc2[4][NTILE] = {};
    for (int ks = 0; ks < 128; ++ks) {
        int krow = (ks * 4 + 2 * half) * 16 + nn;
        v2f bb[NTILE];
#pragma unroll
        for (int nt = 0; nt < NTILE; ++nt) {
            bb[nt].x = hs[nt * PANEL + krow];
            bb[nt].y = hs[nt * PANEL + krow + 16];
        }
        const v2f* ap = (const v2f*)wopk + ((size_t)ks * 32 + wave * 4) * 32 + lane;
#pragma unroll
        for (int mi = 0; mi < 4; ++mi) {
            v2f aa = ap[(size_t)mi * 32];
#pragma unroll
            for (int nt = 0; nt < NTILE; ++nt)
                acc2[mi][nt] = __builtin_amdgcn_wmma_f32_16x16x4_f32(
                    false, aa, false, bb[nt], (short)0, acc2[mi][nt], false, false);
        }
    }
#pragma unroll
    for (int mi = 0; mi < 4; ++mi) {
        int m0 = (wave * 4 + mi) * 16;
#pragma unroll
        for (int v = 0; v < 8; ++v) {
            int o = m0 + v + 8 * half;
            float bias = b_out[o];
            size_t obase = (size_t)b * CPN + (size_t)o * PN + (size_t)p * N_DIM + n0 + nn;
#pragma unroll
            for (int nt = 0; nt < NTILE; ++nt)
                out[obase + nt * 16] = acc2[mi][nt][v] + bias;
        }
    }
}

// ---------------------------------------------------------------------------
extern "C" void kernel_launch(void* const* d_in, const int* in_sizes, int n_in,
                              void* d_out, int out_size, void* d_ws, size_t ws_size,
                              hipStream_t stream) {
    const float* x     = (const float*)d_in[0];
    const float* w_qkv = (const float*)d_in[1];
    const float* b_qkv = (const float*)d_in[2];
    const float* w_out = (const float*)d_in[3];
    const float* b_out = (const float*)d_in[4];
    float* out = (float*)d_out;
    float* ws  = (float*)d_ws;

    float* q     = ws + WS_Q;
    float* score = ws + WS_SCORE;
    float* xbar  = ws + WS_XBAR;
    float* cv    = ws + WS_CV;
    float* wvpk  = ws + WS_WVPK;
    float* wopk  = ws + WS_WOPK;

    const int smem_main = 2 * NTILE * PANEL * (int)sizeof(float);   // 256 KB (<= 320 KB/WGP)
    (void)hipFuncSetAttribute((const void*)main_kernel,
                              hipFuncAttributeMaxDynamicSharedMemorySize, smem_main);

    pack_w_kernel <<<512,  256, 0, stream>>>(w_qkv, w_out, wvpk, wopk);
    q_kernel      <<<256,  256, 0, stream>>>(x, w_qkv, b_qkv, q);
    softmax_kernel<<<64,   256, 0, stream>>>(q, score);
    xbar_kernel   <<<4096, 256, 0, stream>>>(x, score, xbar);
    cv_kernel     <<<4096, 256, 0, stream>>>(w_qkv, b_qkv, xbar, cv);
    main_kernel   <<<1024, 256, smem_main, stream>>>(x, b_qkv, b_out, cv, wvpk, wopk, out);
}